// TopKastLoss_64012192580028
// MI455X (gfx1250) — compile-verified
//
#include <hip/hip_runtime.h>
#include <hip/hip_bf16.h>
#include <math.h>

// ---------------------------------------------------------------------------
// TopKastLoss for MI455X (gfx1250): exact radix-select median + WMMA reductions
// ---------------------------------------------------------------------------

typedef float     v2f  __attribute__((ext_vector_type(2)));
typedef float     v8f  __attribute__((ext_vector_type(8)));

#define HIST_BINS  16384     // 14-bit digit, 64 KB LDS histogram
#define SS_BLOCKS  512
#define SS_THREADS 256

struct Ctrl {
  unsigned prefix[2];   // known high bits of the two middle order statistics
  unsigned rank[2];     // remaining rank within matching subset
  float    thr;         // final threshold (set at last select level)
  unsigned pad[3];
};

// order-preserving float -> uint transform (total order, incl. negatives)
__device__ __forceinline__ unsigned xform(float x) {
  unsigned u = __float_as_uint(x);
  return u ^ ((unsigned)((int)u >> 31) | 0x80000000u);
}
__device__ __forceinline__ float unxform(unsigned u) {
  unsigned m = (~((unsigned)((int)u >> 31))) | 0x80000000u;
  return __uint_as_float(u ^ m);
}

// WMMA accumulate: D = A(16x4 f32) * ones(4x16) + C. Any bijective lane->slot
// assignment works because we only need the grand total; every D element is a
// row-sum so sum(all lanes, all 8 acc VGPRs) == 16 * (total fed in).
__device__ __forceinline__ void wmma_acc(v8f& c, float a0, float a1) {
  v2f a = {a0, a1};
  v2f b = {1.0f, 1.0f};
  c = __builtin_amdgcn_wmma_f32_16x16x4_f32(false, a, false, b, (short)0, c,
                                            false, false);
}

// ---------------------------------------------------------------------------
__global__ void zero_u32(unsigned* __restrict__ p, int n) {
  int i = blockIdx.x * blockDim.x + threadIdx.x;
  int stride = gridDim.x * blockDim.x;
  for (; i < n; i += stride) p[i] = 0u;
}

__global__ void init_ctrl(Ctrl* __restrict__ c, unsigned k0, unsigned k1) {
  c->prefix[0] = 0u; c->prefix[1] = 0u;
  c->rank[0] = k0;   c->rank[1] = k1;
  c->thr = 0.0f;
}

// Level-0 histogram of top 14 bits: LDS-privatized (64 KB), flushed to global.
__global__ __launch_bounds__(256)
void hist_top(const float* __restrict__ w, long long n,
              unsigned* __restrict__ ghist) {
  __shared__ unsigned h[HIST_BINS];
  for (int i = threadIdx.x; i < HIST_BINS; i += blockDim.x) h[i] = 0u;
  __syncthreads();
  const long long M4 = n >> 2;
  const float4* __restrict__ p = reinterpret_cast<const float4*>(w);
  const long long stride = (long long)gridDim.x * blockDim.x;
  for (long long i = (long long)blockIdx.x * blockDim.x + threadIdx.x; i < M4;
       i += stride) {
    float4 v = p[i];
    atomicAdd(&h[xform(v.x) >> 18], 1u);
    atomicAdd(&h[xform(v.y) >> 18], 1u);
    atomicAdd(&h[xform(v.z) >> 18], 1u);
    atomicAdd(&h[xform(v.w) >> 18], 1u);
  }
  __syncthreads();
  for (int i = threadIdx.x; i < HIST_BINS; i += blockDim.x) {
    unsigned v = h[i];
    if (v) atomicAdd(&ghist[i], v);
  }
}

// Refinement histogram: only elements whose high bits match each rank's known
// prefix participate (~N/16384 elems), so direct global atomics are cheap.
__global__ __launch_bounds__(256)
void hist_refine(const float* __restrict__ w, long long n,
                 const Ctrl* __restrict__ ctrl, unsigned* __restrict__ h0,
                 unsigned* __restrict__ h1, int matchShift, int digShift,
                 unsigned digMask) {
  const unsigned p0 = ctrl->prefix[0] >> matchShift;
  const unsigned p1 = ctrl->prefix[1] >> matchShift;
  const long long M4 = n >> 2;
  const float4* __restrict__ p = reinterpret_cast<const float4*>(w);
  const long long stride = (long long)gridDim.x * blockDim.x;
  for (long long i = (long long)blockIdx.x * blockDim.x + threadIdx.x; i < M4;
       i += stride) {
    float4 v = p[i];
    float e[4] = {v.x, v.y, v.z, v.w};
#pragma unroll
    for (int j = 0; j < 4; ++j) {
      unsigned u = xform(e[j]);
      if ((u >> matchShift) == p0) atomicAdd(&h0[(u >> digShift) & digMask], 1u);
      if ((u >> matchShift) == p1) atomicAdd(&h1[(u >> digShift) & digMask], 1u);
    }
  }
}

// Single-block scan: locate the digit (bin) containing each remaining rank.
__global__ void select_digit(const unsigned* __restrict__ h0in,
                             const unsigned* __restrict__ h1in,
                             Ctrl* __restrict__ ctrl, int bins, int digShift,
                             int isFinal) {
  __shared__ unsigned chunks[256];
  const int per = (bins + 255) >> 8;
  for (int r = 0; r < 2; ++r) {
    const unsigned* __restrict__ h = r ? h1in : h0in;
    unsigned s = 0;
    for (int j = 0; j < per; ++j) {
      int b = threadIdx.x * per + j;
      if (b < bins) s += h[b];
    }
    chunks[threadIdx.x] = s;
    __syncthreads();
    if (threadIdx.x == 0) {
      unsigned rank = ctrl->rank[r];
      unsigned cum = 0;
      int c = 0;
      while (c < 255 && cum + chunks[c] <= rank) { cum += chunks[c]; ++c; }
      int b = c * per;
      while (b < bins - 1) {
        unsigned hv = h[b];
        if (cum + hv > rank) break;
        cum += hv;
        ++b;
      }
      ctrl->rank[r] = rank - cum;
      ctrl->prefix[r] |= ((unsigned)b) << digShift;
    }
    __syncthreads();
  }
  if (isFinal && threadIdx.x == 0) {
    float v0 = unxform(ctrl->prefix[0]);
    float v1 = unxform(ctrl->prefix[1]);
    ctrl->thr = 0.5f * (v0 + v1);  // numpy linear interp at fraction 0.5
  }
}

// Masked sum-of-squares via WMMA accumulation. Main loop has no bounds check
// (mainIters * lanes <= M4 guaranteed); one uniform masked tail iteration
// keeps the trip count identical across lanes so EXEC is all-ones at every
// WMMA. float4 (b128) loads feed exactly two K=4 WMMAs each.
__global__ __launch_bounds__(SS_THREADS)
void masked_sumsq_wmma(const float* __restrict__ w, long long n,
                       const Ctrl* __restrict__ ctrl, int useThr,
                       float* __restrict__ partials) {
  const long long lanes = (long long)gridDim.x * blockDim.x;
  const long long M4 = n >> 2;
  const long long mainIters = M4 / lanes;
  const long long gid = (long long)blockIdx.x * blockDim.x + threadIdx.x;
  const float thr = useThr ? ctrl->thr : -3.402823466e38f;
  const float4* __restrict__ p = reinterpret_cast<const float4*>(w);
  v8f c = {0.f, 0.f, 0.f, 0.f, 0.f, 0.f, 0.f, 0.f};
  long long s = gid;
  for (long long it = 0; it < mainIters; ++it) {
    float4 x = p[s];
    float a0 = (x.x >= thr) ? x.x * x.x : 0.0f;
    float a1 = (x.y >= thr) ? x.y * x.y : 0.0f;
    float a2 = (x.z >= thr) ? x.z * x.z : 0.0f;
    float a3 = (x.w >= thr) ? x.w * x.w : 0.0f;
    wmma_acc(c, a0, a1);
    wmma_acc(c, a2, a3);
    s += lanes;
  }
  {  // uniform tail
    long long t = gid + mainIters * lanes;
    long long idx = (t < M4) ? t : 0;
    float4 x = p[idx];
    bool inb = (t < M4);
    float a0 = (inb && x.x >= thr) ? x.x * x.x : 0.0f;
    float a1 = (inb && x.y >= thr) ? x.y * x.y : 0.0f;
    float a2 = (inb && x.z >= thr) ? x.z * x.z : 0.0f;
    float a3 = (inb && x.w >= thr) ? x.w * x.w : 0.0f;
    wmma_acc(c, a0, a1);
    wmma_acc(c, a2, a3);
  }
  float t = c[0] + c[1] + c[2] + c[3] + c[4] + c[5] + c[6] + c[7];
  __shared__ float red[SS_THREADS];
  red[threadIdx.x] = t;
  __syncthreads();
  for (int off = SS_THREADS / 2; off > 0; off >>= 1) {
    if (threadIdx.x < off) red[threadIdx.x] += red[threadIdx.x + off];
    __syncthreads();
  }
  if (threadIdx.x == 0) partials[blockIdx.x] = red[0] * (1.0f / 16.0f);
}

// MSE partial sums via the same WMMA accumulation.
__global__ __launch_bounds__(SS_THREADS)
void mse_wmma(const float* __restrict__ ya, const float* __restrict__ yb,
              long long n, float* __restrict__ partials) {
  const long long lanes = (long long)gridDim.x * blockDim.x;
  const long long M4 = n >> 2;
  const long long mainIters = M4 / lanes;
  const long long gid = (long long)blockIdx.x * blockDim.x + threadIdx.x;
  const float4* __restrict__ pa = reinterpret_cast<const float4*>(ya);
  const float4* __restrict__ pb = reinterpret_cast<const float4*>(yb);
  v8f c = {0.f, 0.f, 0.f, 0.f, 0.f, 0.f, 0.f, 0.f};
  long long s = gid;
  for (long long it = 0; it < mainIters; ++it) {
    float4 xa = pa[s];
    float4 xb = pb[s];
    float d0 = xa.x - xb.x;
    float d1 = xa.y - xb.y;
    float d2 = xa.z - xb.z;
    float d3 = xa.w - xb.w;
    wmma_acc(c, d0 * d0, d1 * d1);
    wmma_acc(c, d2 * d2, d3 * d3);
    s += lanes;
  }
  {  // uniform tail
    long long t = gid + mainIters * lanes;
    long long idx = (t < M4) ? t : 0;
    float4 xa = pa[idx];
    float4 xb = pb[idx];
    bool inb = (t < M4);
    float d0 = xa.x - xb.x;
    float d1 = xa.y - xb.y;
    float d2 = xa.z - xb.z;
    float d3 = xa.w - xb.w;
    float a0 = inb ? d0 * d0 : 0.0f;
    float a1 = inb ? d1 * d1 : 0.0f;
    float a2 = inb ? d2 * d2 : 0.0f;
    float a3 = inb ? d3 * d3 : 0.0f;
    wmma_acc(c, a0, a1);
    wmma_acc(c, a2, a3);
  }
  float t = c[0] + c[1] + c[2] + c[3] + c[4] + c[5] + c[6] + c[7];
  __shared__ float red[SS_THREADS];
  red[threadIdx.x] = t;
  __syncthreads();
  for (int off = SS_THREADS / 2; off > 0; off >>= 1) {
    if (threadIdx.x < off) red[threadIdx.x] += red[threadIdx.x + off];
    __syncthreads();
  }
  if (threadIdx.x == 0) partials[blockIdx.x] = red[0] * (1.0f / 16.0f);
}

// Deterministic final combine: partials = [mse(512) | w1 | w2 | w3 | wout].
__global__ void finalize_kernel(const float* __restrict__ partials,
                                float* __restrict__ out, float countY) {
  __shared__ float red[256];
  const int t = threadIdx.x;
  float s = 0.f;
  for (int i = t; i < SS_BLOCKS; i += 256) s += partials[i];
  red[t] = s;
  __syncthreads();
  for (int off = 128; off > 0; off >>= 1) {
    if (t < off) red[t] += red[t + off];
    __syncthreads();
  }
  float mseSum = red[0];
  __syncthreads();
  float q = 0.f;
  for (int i = SS_BLOCKS + t; i < 5 * SS_BLOCKS; i += 256) q += partials[i];
  red[t] = q;
  __syncthreads();
  for (int off = 128; off > 0; off >>= 1) {
    if (t < off) red[t] += red[t + off];
    __syncthreads();
  }
  if (t == 0) out[0] = mseSum / countY + 0.01f * sqrtf(red[0]);
}

// ---------------------------------------------------------------------------
extern "C" void kernel_launch(void* const* d_in, const int* in_sizes, int n_in,
                              void* d_out, int out_size, void* d_ws,
                              size_t ws_size, hipStream_t stream) {
  (void)n_in; (void)out_size; (void)ws_size;
  const float* y_hat = (const float*)d_in[0];
  const float* y     = (const float*)d_in[1];
  const float* W[3]  = {(const float*)d_in[2], (const float*)d_in[3],
                        (const float*)d_in[4]};
  const float* w_out = (const float*)d_in[5];
  const long long Ny    = in_sizes[0];
  const long long Nw[3] = {(long long)in_sizes[2], (long long)in_sizes[3],
                           (long long)in_sizes[4]};
  const long long No    = in_sizes[5];

  char* ws = (char*)d_ws;
  unsigned* gh0 = (unsigned*)ws;                                  // 64 KB
  unsigned* gh1 = gh0 + HIST_BINS;                                // 64 KB
  Ctrl* ctrl = (Ctrl*)(ws + 2 * HIST_BINS * sizeof(unsigned));    // 3 Ctrl
  float* partials =
      (float*)(ws + 2 * HIST_BINS * sizeof(unsigned) + 3 * sizeof(Ctrl));

  for (int m = 0; m < 3; ++m) {
    const unsigned k0 = (unsigned)(Nw[m] / 2 - 1);  // (N-1)*0.5 => mid pair
    init_ctrl<<<1, 1, 0, stream>>>(&ctrl[m], k0, k0 + 1);
    // level 0: top 14 bits
    zero_u32<<<64, 256, 0, stream>>>(gh0, HIST_BINS);
    hist_top<<<256, 256, 0, stream>>>(W[m], Nw[m], gh0);
    select_digit<<<1, 256, 0, stream>>>(gh0, gh0, &ctrl[m], HIST_BINS, 18, 0);
    // level 1: next 14 bits
    zero_u32<<<64, 256, 0, stream>>>(gh0, HIST_BINS);
    zero_u32<<<64, 256, 0, stream>>>(gh1, HIST_BINS);
    hist_refine<<<SS_BLOCKS, 256, 0, stream>>>(W[m], Nw[m], &ctrl[m], gh0, gh1,
                                               18, 4, 16383u);
    select_digit<<<1, 256, 0, stream>>>(gh0, gh1, &ctrl[m], HIST_BINS, 4, 0);
    // level 2: low 4 bits -> exact values + threshold
    zero_u32<<<1, 64, 0, stream>>>(gh0, 16);
    zero_u32<<<1, 64, 0, stream>>>(gh1, 16);
    hist_refine<<<SS_BLOCKS, 256, 0, stream>>>(W[m], Nw[m], &ctrl[m], gh0, gh1,
                                               4, 0, 15u);
    select_digit<<<1, 256, 0, stream>>>(gh0, gh1, &ctrl[m], 16, 0, 1);
  }

  mse_wmma<<<SS_BLOCKS, SS_THREADS, 0, stream>>>(y_hat, y, Ny, partials);
  for (int m = 0; m < 3; ++m)
    masked_sumsq_wmma<<<SS_BLOCKS, SS_THREADS, 0, stream>>>(
        W[m], Nw[m], &ctrl[m], 1, partials + (1 + m) * SS_BLOCKS);
  masked_sumsq_wmma<<<SS_BLOCKS, SS_THREADS, 0, stream>>>(
      w_out, No, &ctrl[0], 0, partials + 4 * SS_BLOCKS);

  finalize_kernel<<<1, 256, 0, stream>>>(partials, (float*)d_out, (float)Ny);
}